// HyperConvCausal_81870666597033
// MI455X (gfx1250) — compile-verified
//
#include <hip/hip_runtime.h>

typedef __attribute__((ext_vector_type(2))) float v2f;
typedef __attribute__((ext_vector_type(8))) float v8f;

constexpr int IN_CH = 64, OUT_CH = 64, KS = 3, DIL = 2, ZDIM = 128, HID = 32;
constexpr int NB = 16, T = 16384, K = 64, CHUNK = 256;
constexpr int CKS  = IN_CH * KS;      // 192 unfolded channels
constexpr int WELEM = CKS * OUT_CH;   // 12288 weights per (b,k)
constexpr int WT_S = 196;             // LDS stride Wt[o][a]: even (b64 ok), 196%64=4 -> conflict-free
constexpr int HT_S = 34;             // LDS stride hsT[k][h]: even, 34%64 -> lanes hit distinct bank pairs
constexpr int LD_S = 66;             // LDS stride ldsD[c][k]: 8 rows -> +528 -> bank+16, halves disjoint

__device__ __forceinline__ v8f wmma_f32_16x16x4(v2f a, v2f b, v8f c) {
  // D = A(16x4 f32) * B(4x16 f32) + C(16x16 f32), wave32
  return __builtin_amdgcn_wmma_f32_16x16x4_f32(false, a, false, b, (short)0, c, false, false);
}

// ---------------------------------------------------------------------------
// Stage 1a: per (b,k) hidden activations h, hb and output bias.
// 1024 threads total (4 blocks x 256). h/hb kept in registers (unrolled).
// ---------------------------------------------------------------------------
__global__ __launch_bounds__(256) void hyper_stage1a(
    const float* __restrict__ z,   const float* __restrict__ w1,  const float* __restrict__ b1,
    const float* __restrict__ bw1, const float* __restrict__ bb1,
    const float* __restrict__ bw2, const float* __restrict__ bb2,
    float* __restrict__ hmatG,     // [NB][HID][K]
    float* __restrict__ wsB)       // [NB][OUT_CH][K]
{
  const int tg = blockIdx.x * 256 + threadIdx.x;   // 0..1023
  const int b = tg >> 6, k = tg & 63;

  float h[HID], hb[HID];
#pragma unroll
  for (int hh = 0; hh < HID; ++hh) { h[hh] = b1[hh]; hb[hh] = bb1[hh]; }

#pragma unroll 4
  for (int i = 0; i < ZDIM; ++i) {
    const float zi = z[(size_t)(b * ZDIM + i) * K + k];
#pragma unroll
    for (int hh = 0; hh < HID; ++hh) h[hh]  = fmaf(w1[hh * ZDIM + i],  zi, h[hh]);
#pragma unroll
    for (int hh = 0; hh < HID; ++hh) hb[hh] = fmaf(bw1[hh * ZDIM + i], zi, hb[hh]);
  }
#pragma unroll
  for (int hh = 0; hh < HID; ++hh) {
    h[hh]  = fmaxf(h[hh], 0.f);
    hb[hh] = fmaxf(hb[hh], 0.f);
  }
#pragma unroll
  for (int hh = 0; hh < HID; ++hh)
    hmatG[(size_t)b * HID * K + hh * K + k] = h[hh];

  for (int o = 0; o < OUT_CH; ++o) {
    float s = bb2[o];
#pragma unroll
    for (int hh = 0; hh < HID; ++hh) s = fmaf(bw2[o * HID + hh], hb[hh], s);
    wsB[(size_t)b * OUT_CH * K + o * K + k] = s;
  }
}

// ---------------------------------------------------------------------------
// Stage 1b: wflat[b] = w2[12288,32] @ H[32,64] + b2  via f32 WMMA.
// Grid (96, NB): block = 128 c-rows x all 64 k. H kept transposed in LDS
// (hsT[k][h]) so each B fragment is one contiguous ds_load_b64. Result + b2
// staged in padded LDS and written out transposed so wsW[(b*64+k)*12288 + c]
// stores are coalesced.
// ---------------------------------------------------------------------------
__global__ __launch_bounds__(256) void hyper_wgen_wmma(
    const float* __restrict__ w2, const float* __restrict__ b2,
    const float* __restrict__ hmatG,
    float* __restrict__ wsW)      // [NB*K][WELEM]
{
  __shared__ float hsT[K * HT_S];       // [k][h]
  __shared__ float ldsD[128 * LD_S];    // [c_local][k]

  const int b = blockIdx.y;
  const int cbase = blockIdx.x * 128;
  const int tid = threadIdx.x;
  const int lane = tid & 31, wave = tid >> 5;
  const int l16 = lane & 15, lhalf = lane >> 4;

#pragma unroll
  for (int i = 0; i < 8; ++i) {
    int idx = i * 256 + tid;                      // h*64 + k
    hsT[(idx & 63) * HT_S + (idx >> 6)] = hmatG[(size_t)b * HID * K + idx];
  }
  __syncthreads();

  const int crow = cbase + wave * 16 + l16;       // A-matrix row for this lane
  v8f acc[4];
  const v8f vz = {0.f, 0.f, 0.f, 0.f, 0.f, 0.f, 0.f, 0.f};
#pragma unroll
  for (int nt = 0; nt < 4; ++nt) acc[nt] = vz;

#pragma unroll
  for (int ks = 0; ks < 8; ++ks) {                // K = 32, step 4
    // A: lanes<16 -> {h0,h0+1}, lanes>=16 -> {h0+2,h0+3} for row crow
    const v2f a = *(const v2f*)(w2 + (size_t)crow * HID + ks * 4 + lhalf * 2);
#pragma unroll
    for (int nt = 0; nt < 4; ++nt) {
      // B row pair {h0+2*lhalf, +1} at column nt*16+l16 -> contiguous in hsT
      const v2f bf = *(const v2f*)&hsT[(nt * 16 + l16) * HT_S + ks * 4 + lhalf * 2];
      acc[nt] = wmma_f32_16x16x4(a, bf, acc[nt]);
    }
  }

  // acc[nt][v]: row c_local = wave*16 + v + 8*lhalf, col k = nt*16 + l16
#pragma unroll
  for (int nt = 0; nt < 4; ++nt) {
#pragma unroll
    for (int v = 0; v < 8; ++v) {
      int cl = wave * 16 + v + 8 * lhalf;
      ldsD[cl * LD_S + nt * 16 + l16] = acc[nt][v] + b2[cbase + cl];
    }
  }
  __syncthreads();

#pragma unroll 4
  for (int i = 0; i < 32; ++i) {
    int idx = i * 256 + tid;                      // 8192 = 64 k * 128 c
    int k = idx >> 7, cl = idx & 127;
    wsW[((size_t)(b * K + k)) * WELEM + cbase + cl] = ldsD[cl * LD_S + k];
  }
}

// ---------------------------------------------------------------------------
// Stage 2 core: 3 taps x 16 k-steps x (4 M-tiles x 2 N-tiles) f32 WMMA.
// GUARD=true only for the wave that can touch t<0 (kchunk==0, wave==0).
// kk fully unrolled: x loads become base + immediate-offset (kk*4*T*4 bytes).
// ---------------------------------------------------------------------------
template <bool GUARD>
__device__ __forceinline__ void chunk_gemm_core(
    const float* __restrict__ xb, const float* Wt,
    int t0, int l16, int lhalf, v8f (&acc)[4][2])
{
  for (int tap = 0; tap < KS; ++tap) {
    // Per-(nt,j) base pointers; t is kk-invariant.
    const float* pB[2][2];
    bool ok[2];
#pragma unroll
    for (int nt = 0; nt < 2; ++nt) {
      int t = t0 + nt * 16 + l16 - DIL * tap;
      ok[nt] = (t >= 0);
      int tc = GUARD ? (t < 0 ? 0 : t) : t;
#pragma unroll
      for (int j = 0; j < 2; ++j)
        pB[nt][j] = xb + (size_t)(j + 2 * lhalf) * T + tc;
    }
    const int abase = tap * 64 + lhalf * 2;

#pragma unroll
    for (int kk = 0; kk < 16; ++kk) {
      // A fragments from LDS: row o = mt*16+l16, cols {a0..a0+1}/{a0+2..a0+3}
      v2f afrag[4];
#pragma unroll
      for (int mt = 0; mt < 4; ++mt)
        afrag[mt] = *(const v2f*)&Wt[(mt * 16 + l16) * WT_S + abase + kk * 4];

      // B fragments from global x, immediate offsets across kk
      v2f bfrag[2];
#pragma unroll
      for (int nt = 0; nt < 2; ++nt) {
#pragma unroll
        for (int j = 0; j < 2; ++j) {
          float v = pB[nt][j][kk * 4 * T];
          if (GUARD) v = ok[nt] ? v : 0.f;
          bfrag[nt][j] = v;
        }
      }

#pragma unroll
      for (int mt = 0; mt < 4; ++mt)
#pragma unroll
        for (int nt = 0; nt < 2; ++nt)
          acc[mt][nt] = wmma_f32_16x16x4(afrag[mt], bfrag[nt], acc[mt][nt]);
    }
  }
}

// ---------------------------------------------------------------------------
// Stage 2: per-(b,k) chunk GEMM  D[64,256] = Wt[64,192] * xm^T[192,256] + bias
// Grid (K, NB), 256 threads = 8 waves. Wave w: N cols [w*32, w*32+32).
// ---------------------------------------------------------------------------
__global__ __launch_bounds__(256) void hyper_chunk_gemm(
    const float* __restrict__ x,
    const float* __restrict__ wsW, const float* __restrict__ wsB,
    float* __restrict__ y)
{
  __shared__ float Wt[OUT_CH * WT_S];   // Wt[o][a] = W[a][o]
  __shared__ float bias_s[OUT_CH];

  const int kchunk = blockIdx.x;
  const int b = blockIdx.y;
  const int tid = threadIdx.x;
  const int lane = tid & 31, wave = tid >> 5;
  const int l16 = lane & 15, lhalf = lane >> 4;

  // Load W for this (b,k): coalesced global read, transposed into padded LDS.
  const float* __restrict__ wsrc = wsW + ((size_t)(b * K + kchunk)) * WELEM;
#pragma unroll 4
  for (int j = 0; j < WELEM / 256; ++j) {
    int idx = j * 256 + tid;                      // c = a*64 + o
    float v = wsrc[idx];
    Wt[(idx & 63) * WT_S + (idx >> 6)] = v;
  }
  if (tid < OUT_CH)
    bias_s[tid] = wsB[(size_t)b * OUT_CH * K + tid * K + kchunk];
  __syncthreads();

  const int nbase = wave * 32;                    // two 16-wide N tiles
  const int t0 = kchunk * CHUNK + nbase;
  const float* __restrict__ xb = x + (size_t)b * IN_CH * T;

  v8f acc[4][2];
  const v8f vz = {0.f, 0.f, 0.f, 0.f, 0.f, 0.f, 0.f, 0.f};
#pragma unroll
  for (int mt = 0; mt < 4; ++mt)
#pragma unroll
    for (int nt = 0; nt < 2; ++nt) acc[mt][nt] = vz;

  if (t0 >= DIL * (KS - 1))                       // wave-uniform; EXEC stays full
    chunk_gemm_core<false>(xb, Wt, t0, l16, lhalf, acc);
  else
    chunk_gemm_core<true>(xb, Wt, t0, l16, lhalf, acc);

  // Epilogue: y[b][o][t], coalesced along t; non-temporal (write-once stream).
  //   o = mt*16 + v + 8*lhalf,  n = nbase + nt*16 + l16
  float* __restrict__ yb = y + (size_t)b * OUT_CH * T + (size_t)kchunk * CHUNK;
#pragma unroll
  for (int mt = 0; mt < 4; ++mt) {
#pragma unroll
    for (int nt = 0; nt < 2; ++nt) {
#pragma unroll
      for (int v = 0; v < 8; ++v) {
        int o = mt * 16 + v + 8 * lhalf;
        int n = nbase + nt * 16 + l16;
        __builtin_nontemporal_store(acc[mt][nt][v] + bias_s[o], &yb[(size_t)o * T + n]);
      }
    }
  }
}

// ---------------------------------------------------------------------------
extern "C" void kernel_launch(void* const* d_in, const int* in_sizes, int n_in,
                              void* d_out, int out_size, void* d_ws, size_t ws_size,
                              hipStream_t stream) {
  (void)in_sizes; (void)n_in; (void)out_size; (void)ws_size;
  const float* x   = (const float*)d_in[0];
  const float* z   = (const float*)d_in[1];
  const float* w1  = (const float*)d_in[2];
  const float* b1  = (const float*)d_in[3];
  const float* w2  = (const float*)d_in[4];
  const float* b2  = (const float*)d_in[5];
  const float* bw1 = (const float*)d_in[6];
  const float* bb1 = (const float*)d_in[7];
  const float* bw2 = (const float*)d_in[8];
  const float* bb2 = (const float*)d_in[9];
  float* y = (float*)d_out;

  // Workspace layout (floats): wsW [1024][12288] | hmatG [16][32][64] | wsB [16][64][64]
  float* wsW   = (float*)d_ws;
  float* hmatG = wsW + (size_t)NB * K * WELEM;
  float* wsB   = hmatG + (size_t)NB * HID * K;

  hyper_stage1a<<<dim3(4), dim3(256), 0, stream>>>(z, w1, b1, bw1, bb1, bw2, bb2, hmatG, wsB);
  hyper_wgen_wmma<<<dim3(96, NB), dim3(256), 0, stream>>>(w2, b2, hmatG, wsW);
  hyper_chunk_gemm<<<dim3(K, NB), dim3(256), 0, stream>>>(x, wsW, wsB, y);
}